// HGCN_88768384074092
// MI455X (gfx1250) — compile-verified
//
#include <hip/hip_runtime.h>
#include <math.h>

// ---------------------------------------------------------------------------
// HGCN forward (2 hyperbolic GCN layers, Poincare ball, c = 1)
// n = 4096 nodes, fdim = 256 input features, dim = 128 hidden.
// Heavy op: support = (sigmoid(L_i + R_j + ab) * adj_ij) @ xt
//   -> fused fp32 WMMA GEMM (V_WMMA_F32_16X16X4_F32); adj streamed once per
//      layer (64MB, bandwidth-bound; second layer read hits the 192MB L2).
// LDS layouts tuned so each WMMA fragment is a single aligned ds_load_b64.
// ---------------------------------------------------------------------------

typedef __attribute__((ext_vector_type(2))) float v2f;
typedef __attribute__((ext_vector_type(8))) float v8f;

#define N_NODES 4096
#define FDIM    256
#define DIM     128
#define MAXNORM 0.996f        // (1 - EPS) / sqrt(c)
#define MINV    1e-7f

// ---------------------------- block reductions -----------------------------

__device__ __forceinline__ float blk_sum(float v, float* red) {
    const int t  = threadIdx.x;
    const int nt = blockDim.x;
    red[t] = v; __syncthreads();
    for (int s = nt >> 1; s > 0; s >>= 1) {
        if (t < s) red[t] += red[t + s];
        __syncthreads();
    }
    float r = red[0]; __syncthreads();
    return r;
}

__device__ __forceinline__ float blk_max(float v, float* red) {
    const int t  = threadIdx.x;
    const int nt = blockDim.x;
    red[t] = v; __syncthreads();
    for (int s = nt >> 1; s > 0; s >>= 1) {
        if (t < s) red[t] = fmaxf(red[t], red[t + s]);
        __syncthreads();
    }
    float r = red[0]; __syncthreads();
    return r;
}

__device__ __forceinline__ float artanh_c(float x) {
    float xc = fminf(fmaxf(x, -1.f + 1e-7f), 1.f - 1e-7f);
    return 0.5f * (log1pf(xc) - log1pf(-xc));
}

__device__ __forceinline__ float tanh_clip(float x) {
    return tanhf(fminf(fmaxf(x, -7.f), 7.f));
}

// ------------------------- expmap0 + proj (per row) -------------------------
__global__ void expmap0_proj_kernel(const float* __restrict__ in,
                                    float* __restrict__ out, int d) {
    __shared__ float red[256];
    const int row = blockIdx.x;
    const int t   = threadIdx.x;
    float u  = in[(size_t)row * d + t];
    float sq = blk_sum(u * u, red);
    float un = fmaxf(sqrtf(sq), MINV);
    float v  = (tanh_clip(un) / un) * u;
    float vq = blk_sum(v * v, red);
    float vn = fmaxf(sqrtf(vq), MINV);
    if (vn > MAXNORM) v = v * (MAXNORM / vn);
    out[(size_t)row * d + t] = v;
}

// ------------------- fp32 WMMA GEMM:  mx = h @ W^T  -------------------------
// A(h): [M x K] row major, W: [128 x K] row major, C(mx): [M x 128].
// Block: 128 threads (4 waves). M-tile 64 (16 rows/wave), N = 128, K-chunk 64.
// sA: row-major floats, stride 68 (16B-aligned rows, conflict-free frag reads)
// sB: pair-interleaved v2f: sB[p*LDP+n] = {B[2p][n], B[2p+1][n]}
#define T_LDA 68
#define T_LDP 130

__global__ void gemm_wmma_kernel(const float* __restrict__ A,
                                 const float* __restrict__ W,
                                 float* __restrict__ C, int K) {
    __shared__ float sA[64 * T_LDA];
    __shared__ v2f   sB[32 * T_LDP];

    const int tid    = threadIdx.x;
    const int lane   = tid & 31;
    const int wave   = tid >> 5;
    const int half   = lane >> 4;
    const int lane16 = lane & 15;
    const int m0     = blockIdx.x * 64;
    const int mw     = wave * 16;

    v8f acc[8];
#pragma unroll
    for (int t = 0; t < 8; ++t) acc[t] = (v8f)0.f;

    for (int kc0 = 0; kc0 < K; kc0 += 64) {
        // A tile: 64 rows x 16 float4
        for (int i = tid; i < 64 * 16; i += 128) {
            int kq = i & 15, m = i >> 4;
            float4 v = *(const float4*)(A + (size_t)(m0 + m) * K + kc0 + kq * 4);
            *(float4*)(&sA[m * T_LDA + kq * 4]) = v;
        }
        // B tile: sB[p][n] = {W[n][kc0+2p], W[n][kc0+2p+1]}
        for (int i = tid; i < 128 * 16; i += 128) {
            int kq = i & 15, n = i >> 4;
            float4 v = *(const float4*)(W + (size_t)n * K + kc0 + kq * 4);
            v2f lo; lo.x = v.x; lo.y = v.y;
            v2f hi; hi.x = v.z; hi.y = v.w;
            sB[(2 * kq + 0) * T_LDP + n] = lo;
            sB[(2 * kq + 1) * T_LDP + n] = hi;
        }
        __syncthreads();

#pragma unroll 8
        for (int k0 = 0; k0 < 64; k0 += 4) {
            v2f a = *(const v2f*)(&sA[(mw + lane16) * T_LDA + k0 + 2 * half]);
            const int pb = (k0 >> 1) + half;
#pragma unroll
            for (int t = 0; t < 8; ++t) {
                v2f b = sB[pb * T_LDP + t * 16 + lane16];
                acc[t] = __builtin_amdgcn_wmma_f32_16x16x4_f32(
                    false, a, false, b, (short)0, acc[t], false, false);
            }
        }
        __syncthreads();
    }

#pragma unroll
    for (int t = 0; t < 8; ++t) {
        const int nb = t * 16;
#pragma unroll
        for (int r = 0; r < 8; ++r) {
            int row = m0 + mw + r + 8 * half;
            C[(size_t)row * DIM + nb + lane16] = acc[t][r];
        }
    }
}

// ------------- per-row: matvec finish, bias mobius_add, logmap0 -------------
__global__ void rowops_pre_kernel(const float* __restrict__ hin, int din,
                                  const float* __restrict__ mx,
                                  const float* __restrict__ hbias,
                                  const float* __restrict__ aw,
                                  float* __restrict__ xt,
                                  float* __restrict__ Lv,
                                  float* __restrict__ Rv) {
    __shared__ float red[128];
    const int row = blockIdx.x;
    const int t   = threadIdx.x;

    float s = 0.f;
    for (int j = t; j < din; j += 128) {
        float v = hin[(size_t)row * din + j];
        s += v * v;
    }
    float xn = fmaxf(sqrtf(blk_sum(s, red)), MINV);

    float m    = mx[(size_t)row * DIM + t];
    float mxn  = fmaxf(sqrtf(blk_sum(m * m, red)), MINV);
    float mabs = blk_max(fabsf(m), red);

    float th  = tanh_clip((mxn / xn) * artanh_c(xn));
    float res = th * m / mxn;
    if (mabs == 0.f) res = 0.f;
    float rn = fmaxf(sqrtf(blk_sum(res * res, red)), MINV);
    if (rn > MAXNORM) res = res * (MAXNORM / rn);

    float bv = hbias[t];
    float x2 = blk_sum(res * res, red);
    float y2 = blk_sum(bv * bv, red);
    float xy = blk_sum(res * bv, red);
    float num = (1.f + 2.f * xy + y2) * res + (1.f - x2) * bv;
    float den = fmaxf(1.f + 2.f * xy + x2 * y2, MINV);
    float hb  = num / den;
    float hn = fmaxf(sqrtf(blk_sum(hb * hb, red)), MINV);
    if (hn > MAXNORM) hb = hb * (MAXNORM / hn);
    float pn = fminf(hn, MAXNORM);

    float xv = (artanh_c(pn) / pn) * hb;
    xt[(size_t)row * DIM + t] = xv;

    float ls = blk_sum(xv * aw[t], red);
    float rs = blk_sum(xv * aw[DIM + t], red);
    if (t == 0) { Lv[row] = ls; Rv[row] = rs; }
}

// -------- fused attention aggregation: sup = (sigmoid(..)*adj) @ xt ---------
__global__ void agg_wmma_kernel(const float* __restrict__ adj,
                                const float* __restrict__ xt,
                                const float* __restrict__ Lv,
                                const float* __restrict__ Rv,
                                const float* __restrict__ ab,
                                float* __restrict__ sup, int n) {
    __shared__ float sA[64 * T_LDA];   // att tile [m][k] row-major
    __shared__ v2f   sB[32 * T_LDP];   // xt tile pair-interleaved

    const int tid    = threadIdx.x;
    const int lane   = tid & 31;
    const int wave   = tid >> 5;
    const int half   = lane >> 4;
    const int lane16 = lane & 15;
    const int m0     = blockIdx.x * 64;
    const int mw     = wave * 16;
    const float ab0  = ab[0];

    v8f acc[8];
#pragma unroll
    for (int t = 0; t < 8; ++t) acc[t] = (v8f)0.f;

    for (int kc0 = 0; kc0 < n; kc0 += 64) {
        // A tile: att[m][k] = sigmoid(L[m] + R[k] + ab) * adj[m][k], float4 wide
        for (int i = tid; i < 64 * 16; i += 128) {
            int kq = i & 15, m = i >> 4;
            int gm = m0 + m, gk = kc0 + kq * 4;
            float  lm = Lv[gm];
            float4 rv = *(const float4*)(Rv + gk);
            float4 av = *(const float4*)(adj + (size_t)gm * n + gk);
            float4 o;
            o.x = av.x / (1.f + expf(-(lm + rv.x + ab0)));
            o.y = av.y / (1.f + expf(-(lm + rv.y + ab0)));
            o.z = av.z / (1.f + expf(-(lm + rv.z + ab0)));
            o.w = av.w / (1.f + expf(-(lm + rv.w + ab0)));
            *(float4*)(&sA[m * T_LDA + kq * 4]) = o;
        }
        // B tile: sB[p][n] = {xt[kc0+2p][n], xt[kc0+2p+1][n]}
        for (int i = tid; i < 32 * 32; i += 128) {
            int nq = i & 31, p = i >> 5;
            const float* r0 = xt + (size_t)(kc0 + 2 * p) * DIM + nq * 4;
            float4 a0 = *(const float4*)(r0);
            float4 a1 = *(const float4*)(r0 + DIM);
            v2f* dst = &sB[p * T_LDP + nq * 4];
            v2f t0; t0.x = a0.x; t0.y = a1.x; dst[0] = t0;
            v2f t1; t1.x = a0.y; t1.y = a1.y; dst[1] = t1;
            v2f t2; t2.x = a0.z; t2.y = a1.z; dst[2] = t2;
            v2f t3; t3.x = a0.w; t3.y = a1.w; dst[3] = t3;
        }
        __syncthreads();

#pragma unroll 8
        for (int k0 = 0; k0 < 64; k0 += 4) {
            v2f a = *(const v2f*)(&sA[(mw + lane16) * T_LDA + k0 + 2 * half]);
            const int pb = (k0 >> 1) + half;
#pragma unroll
            for (int t = 0; t < 8; ++t) {
                v2f b = sB[pb * T_LDP + t * 16 + lane16];
                acc[t] = __builtin_amdgcn_wmma_f32_16x16x4_f32(
                    false, a, false, b, (short)0, acc[t], false, false);
            }
        }
        __syncthreads();
    }

#pragma unroll
    for (int t = 0; t < 8; ++t) {
        const int nb = t * 16;
#pragma unroll
        for (int r = 0; r < 8; ++r) {
            int row = m0 + mw + r + 8 * half;
            sup[(size_t)row * DIM + nb + lane16] = acc[t][r];
        }
    }
}

// ---------------- per-row: proj(expmap0), HypAct, output --------------------
__global__ void rowops_post_kernel(const float* __restrict__ sup,
                                   float* __restrict__ out) {
    __shared__ float red[128];
    const int row = blockIdx.x;
    const int t   = threadIdx.x;

    float v  = sup[(size_t)row * DIM + t];
    float un = fmaxf(sqrtf(blk_sum(v * v, red)), MINV);
    float h  = (tanh_clip(un) / un) * v;
    float hn = fmaxf(sqrtf(blk_sum(h * h, red)), MINV);
    if (hn > MAXNORM) h = h * (MAXNORM / hn);
    float pn = fminf(hn, MAXNORM);
    float u  = (artanh_c(pn) / pn) * h;
    u = fmaxf(u, 0.f);
    float un2 = fmaxf(sqrtf(blk_sum(u * u, red)), MINV);
    float w   = (tanh_clip(un2) / un2) * u;
    float wn = fmaxf(sqrtf(blk_sum(w * w, red)), MINV);
    if (wn > MAXNORM) w = w * (MAXNORM / wn);
    out[(size_t)row * DIM + t] = w;
}

// ---------------------------------------------------------------------------

extern "C" void kernel_launch(void* const* d_in, const int* in_sizes, int n_in,
                              void* d_out, int out_size, void* d_ws, size_t ws_size,
                              hipStream_t stream) {
    const float* x    = (const float*)d_in[0];
    const float* adj  = (const float*)d_in[1];
    const float* W1   = (const float*)d_in[2];
    const float* b1   = (const float*)d_in[3];
    const float* aw1  = (const float*)d_in[4];
    const float* ab1  = (const float*)d_in[5];
    const float* W2   = (const float*)d_in[6];
    const float* b2   = (const float*)d_in[7];
    const float* aw2  = (const float*)d_in[8];
    const float* ab2  = (const float*)d_in[9];

    float* ws  = (float*)d_ws;
    float* h   = ws;                             // N*FDIM (layer2 reuses first N*DIM)
    float* mx  = h   + (size_t)N_NODES * FDIM;   // N*DIM
    float* xt  = mx  + (size_t)N_NODES * DIM;    // N*DIM
    float* Lv  = xt  + (size_t)N_NODES * DIM;    // N
    float* Rv  = Lv  + N_NODES;                  // N
    float* sup = Rv  + N_NODES;                  // N*DIM
    float* hb  = sup + (size_t)N_NODES * DIM;    // DIM

    const dim3 rowsD(N_NODES), tiles(N_NODES / 64);

    expmap0_proj_kernel<<<N_NODES, FDIM, 0, stream>>>(x, h, FDIM);

    // ---- layer 1 (din = 256) ----
    expmap0_proj_kernel<<<1, DIM, 0, stream>>>(b1, hb, DIM);
    gemm_wmma_kernel<<<tiles, 128, 0, stream>>>(h, W1, mx, FDIM);
    rowops_pre_kernel<<<rowsD, DIM, 0, stream>>>(h, FDIM, mx, hb, aw1, xt, Lv, Rv);
    agg_wmma_kernel<<<tiles, 128, 0, stream>>>(adj, xt, Lv, Rv, ab1, sup, N_NODES);
    rowops_post_kernel<<<rowsD, DIM, 0, stream>>>(sup, h);

    // ---- layer 2 (din = 128) ----
    expmap0_proj_kernel<<<1, DIM, 0, stream>>>(b2, hb, DIM);
    gemm_wmma_kernel<<<tiles, 128, 0, stream>>>(h, W2, mx, DIM);
    rowops_pre_kernel<<<rowsD, DIM, 0, stream>>>(h, DIM, mx, hb, aw2, xt, Lv, Rv);
    agg_wmma_kernel<<<tiles, 128, 0, stream>>>(adj, xt, Lv, Rv, ab2, sup, N_NODES);
    rowops_post_kernel<<<rowsD, DIM, 0, stream>>>(sup, (float*)d_out);
}